// KernelImagePosterior_71373766525227
// MI455X (gfx1250) — compile-verified
//
#include <hip/hip_runtime.h>
#include <math.h>

// Problem constants (from reference)
constexpr int kB    = 256;   // batch
constexpr int kDIN  = 512;
constexpr int kH    = 512;
constexpr int kDOUT = 10;
constexpr int kS    = 32;    // posterior samples
constexpr int kN    = kB * kDOUT;   // 2560 = output-space vector length
constexpr int kIter = 10;           // fixed CG steps
constexpr float kEPS = 1e-12f;
// per-sample global scratch: V[256x512] + W[256x512] + basis[11 x 2560]
constexpr int kSamp = kB * kH * 2 + (kIter + 1) * kN;

typedef __attribute__((ext_vector_type(2))) float v2f;
typedef __attribute__((ext_vector_type(8))) float v8f;

#define WMMA_F32X4(a, b, c) \
  __builtin_amdgcn_wmma_f32_16x16x4_f32(false, (a), false, (b), (short)0, (c), false, false)

// ---------------------------------------------------------------------------
// WMMA f32 16x16x4 tile helpers (CDNA5 wave32 layouts, ISA 7.12.2)
//   A 16x4 : lane&15 -> M ; VGPR0/1 hold K = 2*(lane>>4) , +1
//   B 4x16 : lane&15 -> N ; VGPR0/1 hold K = 2*(lane>>4) , +1
//   C/D    : VGPR r -> M = r + 8*(lane>>4), N = lane&15
// ---------------------------------------------------------------------------
__device__ inline v8f wmma_tile_ab(const float* __restrict__ A, int lda,
                                   const float* __restrict__ Bm, int ldb,
                                   int K, v8f c) {
  const int lane = threadIdx.x & 31;
  const int mn = lane & 15;
  const int kb = (lane >> 4) << 1;
  for (int k0 = 0; k0 < K; k0 += 4) {
    v2f a, b;
    a.x = A[mn * lda + (k0 + kb)];
    a.y = A[mn * lda + (k0 + kb + 1)];
    b.x = Bm[(k0 + kb) * ldb + mn];
    b.y = Bm[(k0 + kb + 1) * ldb + mn];
    c = WMMA_F32X4(a, b, c);
  }
  return c;
}

// B supplied transposed: B[k][n] = BT[n*ldbt + k]  (for X @ X^T grams)
__device__ inline v8f wmma_tile_abt(const float* __restrict__ A, int lda,
                                    const float* __restrict__ BT, int ldbt,
                                    int K, v8f c) {
  const int lane = threadIdx.x & 31;
  const int mn = lane & 15;
  const int kb = (lane >> 4) << 1;
  for (int k0 = 0; k0 < K; k0 += 4) {
    v2f a, b;
    a.x = A[mn * lda + (k0 + kb)];
    a.y = A[mn * lda + (k0 + kb + 1)];
    b.x = BT[mn * ldbt + (k0 + kb)];
    b.y = BT[mn * ldbt + (k0 + kb + 1)];
    c = WMMA_F32X4(a, b, c);
  }
  return c;
}

__device__ inline void wmma_store(float* __restrict__ D, int ldd, v8f c,
                                  float addv) {
  const int lane = threadIdx.x & 31;
  const int n = lane & 15;
  const int mo = (lane >> 4) << 3;
#pragma unroll
  for (int r = 0; r < 8; ++r) D[(mo + r) * ldd + n] = c[r] + addv;
}

// ---------------------------------------------------------------------------
// Precompute kernels
// ---------------------------------------------------------------------------
// h = tanh(x@W1 + b1), G = 1 - h^2.  grid 512 (16 x 32 tiles), block 32 (1 wave)
__global__ void k_hG(const float* __restrict__ x, const float* __restrict__ W1,
                     const float* __restrict__ b1, float* __restrict__ h,
                     float* __restrict__ G) {
  const int bid = blockIdx.x;
  const int mt = bid >> 5, nt = bid & 31;
  v8f c = {};
  c = wmma_tile_ab(x + mt * 16 * kDIN, kDIN, W1 + nt * 16, kH, kDIN, c);
  const int lane = threadIdx.x & 31;
  const int n = lane & 15;
  const int mo = (lane >> 4) << 3;
#pragma unroll
  for (int r = 0; r < 8; ++r) {
    const int row = mt * 16 + mo + r;
    const int col = nt * 16 + n;
    const float z = c[r] + b1[col];
    const float hv = tanhf(z);
    const int id = row * kH + col;
    h[id] = hv;
    G[id] = 1.f - hv * hv;
  }
}

// out = M @ M^T + ones (B x B Gram).  grid 256 (16x16 tiles), block 32
__global__ void k_gram(const float* __restrict__ M, int ldm, int K,
                       float* __restrict__ out, int ldo) {
  const int bid = blockIdx.x;
  const int mt = bid >> 4, nt = bid & 15;
  v8f c = {};
  c = wmma_tile_abt(M + mt * 16 * ldm, ldm, M + nt * 16 * ldm, ldm, K, c);
  wmma_store(out + mt * 16 * ldo + nt * 16, ldo, c, 1.0f);
}

// out0 = h @ W2 + b2.  One thread per batch row, 10 accumulators in regs.
__global__ void k_out0(const float* __restrict__ h, const float* __restrict__ W2,
                       const float* __restrict__ b2, float* __restrict__ out0) {
  const int b = threadIdx.x;  // 256 threads
  float acc[kDOUT];
#pragma unroll
  for (int o = 0; o < kDOUT; ++o) acc[o] = b2[o];
  const float* hr = h + b * kH;
  for (int j = 0; j < kH; ++j) {
    const float hv = hr[j];
#pragma unroll
    for (int o = 0; o < kDOUT; ++o) acc[o] += hv * W2[j * kDOUT + o];
  }
#pragma unroll
  for (int o = 0; o < kDOUT; ++o) out0[b * kDOUT + o] = acc[o];
}

// Per sample: dh_s = G ⊙ (x @ eps_W1_s + eps_b1_s)  -> sample V scratch
// grid (512 tiles, S), block 32
__global__ void k_dh(const float* __restrict__ x,
                     const float* __restrict__ eps_W1,
                     const float* __restrict__ eps_b1,
                     const float* __restrict__ G, float* __restrict__ samp) {
  const int s = blockIdx.y;
  const int bid = blockIdx.x;
  const int mt = bid >> 5, nt = bid & 31;
  const float* eW1 = eps_W1 + (size_t)s * kDIN * kH;
  v8f c = {};
  c = wmma_tile_ab(x + mt * 16 * kDIN, kDIN, eW1 + nt * 16, kH, kDIN, c);
  float* dh = samp + (size_t)s * kSamp;  // reuse V region as dh staging
  const int lane = threadIdx.x & 31;
  const int n = lane & 15;
  const int mo = (lane >> 4) << 3;
#pragma unroll
  for (int r = 0; r < 8; ++r) {
    const int row = mt * 16 + mo + r;
    const int col = nt * 16 + n;
    const int id = row * kH + col;
    const float dz = c[r] + eps_b1[(size_t)s * kH + col];
    dh[id] = G[id] * dz;
  }
}

// b_s = dh_s @ W2 + h @ eps_W2_s + eps_b2_s.  grid (S) x block 256: one thread
// per batch row, 10 accumulators — each dh/h element read exactly once.
__global__ void k_ball(const float* __restrict__ samp, const float* __restrict__ h,
                       const float* __restrict__ W2,
                       const float* __restrict__ eps_W2,
                       const float* __restrict__ eps_b2,
                       float* __restrict__ b_all) {
  const int s = blockIdx.x;
  const int b = threadIdx.x;  // 256 threads
  const float* dh = samp + (size_t)s * kSamp + b * kH;
  const float* hr = h + b * kH;
  const float* eW2 = eps_W2 + (size_t)s * kH * kDOUT;
  float acc[kDOUT];
#pragma unroll
  for (int o = 0; o < kDOUT; ++o) acc[o] = eps_b2[(size_t)s * kDOUT + o];
  for (int j = 0; j < kH; ++j) {
    const float dv = dh[j];
    const float hv = hr[j];
#pragma unroll
    for (int o = 0; o < kDOUT; ++o)
      acc[o] += dv * W2[j * kDOUT + o] + hv * eW2[j * kDOUT + o];
  }
#pragma unroll
  for (int o = 0; o < kDOUT; ++o)
    b_all[(size_t)s * kN + b * kDOUT + o] = acc[o];
}

// ---------------------------------------------------------------------------
// CG kernel: one workgroup (512 threads = 16 waves) per sample
// matvec(U) = (G ⊙ (Kx (G ⊙ (U W2^T)))) W2 + Kh U
// ---------------------------------------------------------------------------
__device__ inline float block_dot(const float* __restrict__ u,
                                  const float* __restrict__ v,
                                  float* __restrict__ red) {
  const int tid = threadIdx.x;
  float acc = 0.f;
  for (int i = tid; i < kN; i += 512) acc += u[i] * v[i];
  red[tid] = acc;
  __syncthreads();
  for (int off = 256; off > 0; off >>= 1) {
    if (tid < off) red[tid] += red[tid + off];
    __syncthreads();
  }
  const float r = red[0];
  __syncthreads();
  return r;
}

__device__ void cg_matvec(const float* __restrict__ u, float* __restrict__ outv,
                          float* __restrict__ spart,
                          float* __restrict__ V, float* __restrict__ Wm,
                          const float* __restrict__ G,
                          const float* __restrict__ Kx,
                          const float* __restrict__ Kh,
                          const float* __restrict__ W2) {
  const int tid = threadIdx.x;
  // ---- step 1: V = G ⊙ (U @ W2^T)   [256 x 512]
  for (int i = tid; i < kB * kH; i += 512) {
    const int b = i >> 9;
    const int j = i & (kH - 1);
    float acc = 0.f;
#pragma unroll
    for (int o = 0; o < kDOUT; ++o) acc += u[b * kDOUT + o] * W2[j * kDOUT + o];
    V[i] = G[i] * acc;
  }
  __syncthreads();
  // ---- step 2: Wm = Kx @ V via WMMA, register-blocked 16x64 per wave:
  // one A fragment feeds 4 independent WMMAs per k-step (A traffic /4, ILP x4)
  {
    const int wave = tid >> 5;
    const int lane = threadIdx.x & 31;
    const int mn = lane & 15;
    const int kb = (lane >> 4) << 1;
    for (int t = wave; t < 16 * 8; t += 16) {   // 16 mt x 8 groups of 4 n-tiles
      const int mt = t >> 3;
      const int nc0 = (t & 7) << 6;             // starting column (4*16)
      const float* Arow = Kx + (mt * 16 + mn) * kB;
      v8f c0 = {}, c1 = {}, c2 = {}, c3 = {};
      for (int k0 = 0; k0 < kB; k0 += 4) {
        v2f a, b0, b1, b2, b3;
        a.x = Arow[k0 + kb];
        a.y = Arow[k0 + kb + 1];
        const float* Br0 = V + (k0 + kb) * kH + nc0 + mn;
        const float* Br1 = V + (k0 + kb + 1) * kH + nc0 + mn;
        b0.x = Br0[0];  b0.y = Br1[0];
        b1.x = Br0[16]; b1.y = Br1[16];
        b2.x = Br0[32]; b2.y = Br1[32];
        b3.x = Br0[48]; b3.y = Br1[48];
        c0 = WMMA_F32X4(a, b0, c0);
        c1 = WMMA_F32X4(a, b1, c1);
        c2 = WMMA_F32X4(a, b2, c2);
        c3 = WMMA_F32X4(a, b3, c3);
      }
      float* Drow = Wm + mt * 16 * kH + nc0;
      wmma_store(Drow,      kH, c0, 0.f);
      wmma_store(Drow + 16, kH, c1, 0.f);
      wmma_store(Drow + 32, kH, c2, 0.f);
      wmma_store(Drow + 48, kH, c3, 0.f);
    }
  }
  __syncthreads();
  // ---- step 3: out = (G ⊙ Wm) @ W2 + Kh @ U
  // 2 threads per batch row, 10 accumulators in regs: each G/Wm element is
  // read once (was 10x redundant). Halves combined through LDS.
  {
    const int b = tid >> 1;
    const int half = tid & 1;
    float acc[kDOUT];
#pragma unroll
    for (int o = 0; o < kDOUT; ++o) acc[o] = 0.f;
    const float* wr = Wm + b * kH;
    const float* gr = G + b * kH;
    const int j0 = half * (kH / 2);
    for (int j = j0; j < j0 + kH / 2; ++j) {
      const float gw = gr[j] * wr[j];
#pragma unroll
      for (int o = 0; o < kDOUT; ++o) acc[o] += gw * W2[j * kDOUT + o];
    }
    const float* kr = Kh + b * kB;
    const int bb0 = half * (kB / 2);
    for (int bb = bb0; bb < bb0 + kB / 2; ++bb) {
      const float kv = kr[bb];
#pragma unroll
      for (int o = 0; o < kDOUT; ++o) acc[o] += kv * u[bb * kDOUT + o];
    }
    if (half) {
#pragma unroll
      for (int o = 0; o < kDOUT; ++o) spart[b * kDOUT + o] = acc[o];
    }
    __syncthreads();
    if (!half) {
#pragma unroll
      for (int o = 0; o < kDOUT; ++o)
        outv[b * kDOUT + o] = acc[o] + spart[b * kDOUT + o];
    }
  }
  __syncthreads();
}

__global__ __launch_bounds__(512) void k_cg(
    const float* __restrict__ G, const float* __restrict__ Kx,
    const float* __restrict__ Kh, const float* __restrict__ W2,
    const float* __restrict__ out0, const float* __restrict__ b_all,
    const float* __restrict__ log_precision,
    const float* __restrict__ log_scale_image, float* __restrict__ samp_base,
    float* __restrict__ preds) {
  __shared__ float sp[kN], sr[kN], sa[kN], sAp[kN], spart[kN];
  __shared__ float red[512];
  const int s = blockIdx.x;
  const int tid = threadIdx.x;
  float* V = samp_base + (size_t)s * kSamp;
  float* Wm = V + kB * kH;
  float* basis = Wm + kB * kH;
  const float* bvec = b_all + (size_t)s * kN;

  for (int i = tid; i < kN; i += 512) {
    const float bv = bvec[i];
    sp[i] = bv;
    sr[i] = bv;
    sa[i] = 0.f;
  }
  __syncthreads();
  const float rn0 = sqrtf(block_dot(sr, sr, red));
  for (int i = tid; i < kN; i += 512) basis[i] = sr[i] / (rn0 + kEPS);
  __syncthreads();

  float coef[kIter + 1];
  for (int k = 0; k < kIter; ++k) {
    cg_matvec(sp, sAp, spart, V, Wm, G, Kx, Kh, W2);
    const float rr = block_dot(sr, sr, red);
    const float pAp = block_dot(sp, sAp, red);
    const float alpha = rr / (pAp + kEPS);
    for (int i = tid; i < kN; i += 512) {
      sa[i] += alpha * sp[i];
      sr[i] -= alpha * sAp[i];
    }
    __syncthreads();
    // classical Gram-Schmidt: coefficients against unmodified r_new, rows 0..k
    for (int j = 0; j <= k; ++j)
      coef[j] = block_dot(basis + (size_t)j * kN, sr, red);
    for (int i = tid; i < kN; i += 512) {
      float acc = 0.f;
      for (int j = 0; j <= k; ++j) acc += coef[j] * basis[(size_t)j * kN + i];
      sr[i] -= acc;
    }
    __syncthreads();
    const float rr2 = block_dot(sr, sr, red);
    const float beta = rr2 / (rr + kEPS);
    const float rn = sqrtf(rr2);
    for (int i = tid; i < kN; i += 512) {
      sp[i] = sr[i] + beta * sp[i];
      basis[(size_t)(k + 1) * kN + i] = sr[i] / (rn + kEPS);
    }
    __syncthreads();
  }
  // preds = out0 + (sk - si) * (J J^T) a + si * b
  cg_matvec(sa, sAp, spart, V, Wm, G, Kx, Kh, W2);
  const float sk = expf(-0.5f * log_precision[0]);
  const float si = expf(log_scale_image[0]);
  for (int i = tid; i < kN; i += 512)
    preds[(size_t)s * kN + i] = out0[i] + (sk - si) * sAp[i] + si * bvec[i];
}

// ---------------------------------------------------------------------------
extern "C" void kernel_launch(void* const* d_in, const int* in_sizes, int n_in,
                              void* d_out, int out_size, void* d_ws,
                              size_t ws_size, hipStream_t stream) {
  (void)in_sizes; (void)n_in; (void)out_size; (void)ws_size;
  const float* x   = (const float*)d_in[0];
  const float* W1  = (const float*)d_in[1];
  const float* b1  = (const float*)d_in[2];
  const float* W2  = (const float*)d_in[3];
  const float* b2  = (const float*)d_in[4];
  const float* eW1 = (const float*)d_in[5];
  const float* eb1 = (const float*)d_in[6];
  const float* eW2 = (const float*)d_in[7];
  const float* eb2 = (const float*)d_in[8];
  const float* lp  = (const float*)d_in[9];
  const float* lsi = (const float*)d_in[10];

  float* ws    = (float*)d_ws;
  float* h     = ws;                 // 256*512
  float* G     = h + kB * kH;        // 256*512
  float* Kx    = G + kB * kH;        // 256*256
  float* Kh    = Kx + kB * kB;       // 256*256
  float* out0  = Kh + kB * kB;       // 2560
  float* ball  = out0 + kN;          // 32*2560
  float* samp  = ball + kS * kN;     // 32 * kSamp (~37 MB total workspace)
  float* preds = (float*)d_out;      // [32,256,10] f32

  k_hG  <<<dim3(512),     dim3(32),  0, stream>>>(x, W1, b1, h, G);
  k_gram<<<dim3(256),     dim3(32),  0, stream>>>(x, kDIN, kDIN, Kx, kB);
  k_gram<<<dim3(256),     dim3(32),  0, stream>>>(h, kH, kH, Kh, kB);
  k_out0<<<dim3(1),       dim3(256), 0, stream>>>(h, W2, b2, out0);
  k_dh  <<<dim3(512, kS), dim3(32),  0, stream>>>(x, eW1, eb1, G, samp);
  k_ball<<<dim3(kS),      dim3(256), 0, stream>>>(samp, h, W2, eW2, eb2, ball);
  k_cg  <<<dim3(kS),      dim3(512), 0, stream>>>(G, Kx, Kh, W2, out0, ball,
                                                  lp, lsi, samp, preds);
}